// HierarchicalDownsample_39986145526290
// MI455X (gfx1250) — compile-verified
//
#include <hip/hip_runtime.h>
#include <hip/hip_bf16.h>

typedef __attribute__((ext_vector_type(2))) float v2f;
typedef __attribute__((ext_vector_type(8))) float v8f;

#define B_  8
#define N_  16384
#define M_  4096
#define D_  512
#define LDA 516   // padded LDS row stride (floats) -> conflict-free b64 A reads

// ------------------------------------------------------------------
// Kernel 1: farthest point sampling. One 1024-thread block per batch.
// Each thread owns 16 points in registers (strided layout n = k*1024+t).
// ------------------------------------------------------------------
__global__ __launch_bounds__(1024) void fps_kernel(
    const float* __restrict__ coords, const float* __restrict__ times,
    int* __restrict__ idxOut)
{
  const int b    = blockIdx.x;
  const int t    = threadIdx.x;
  const int lane = t & 31;
  const int wv   = t >> 5;

  __shared__ float sval[32];
  __shared__ int   sidx[32];
  __shared__ float slast[3];
  __shared__ int   snext;

  float px[16], py[16], pt[16], dmin[16];
  const float2* c2 = (const float2*)coords;
#pragma unroll
  for (int k = 0; k < 16; ++k) {
    const int n = (k << 10) + t;
    float2 xy = c2[(size_t)b * N_ + n];
    px[k] = xy.x; py[k] = xy.y;
    pt[k] = times[(size_t)b * N_ + n];
    dmin[k] = __builtin_inff();
  }

  if (t == 0) {
    idxOut[(size_t)b * M_] = 0;          // deterministic seed point
    slast[0] = px[0]; slast[1] = py[0]; slast[2] = pt[0];
  }
  __syncthreads();
  float lx = slast[0], ly = slast[1], lt = slast[2];

  for (int s = 1; s < M_; ++s) {
    float best = -__builtin_inff();
    int   bidx = 0;
#pragma unroll
    for (int k = 0; k < 16; ++k) {
      const float dx = px[k] - lx, dy = py[k] - ly, dt = pt[k] - lt;
      const float d  = sqrtf(dx * dx + dy * dy + dt * dt + 1e-8f);
      const float dm = fminf(dmin[k], d);
      dmin[k] = dm;
      const int n = (k << 10) + t;
      if (dm > best) { best = dm; bidx = n; }   // k ascending => lowest n wins ties
    }
    // wave32 argmax reduce, lowest-index tie-break (matches jnp.argmax)
#pragma unroll
    for (int off = 16; off > 0; off >>= 1) {
      const float ov = __shfl_xor(best, off, 32);
      const int   oi = __shfl_xor(bidx, off, 32);
      if (ov > best || (ov == best && oi < bidx)) { best = ov; bidx = oi; }
    }
    if (lane == 0) { sval[wv] = best; sidx[wv] = bidx; }
    __syncthreads();
    if (wv == 0) {
      float v = sval[lane];
      int   i = sidx[lane];
#pragma unroll
      for (int off = 16; off > 0; off >>= 1) {
        const float ov = __shfl_xor(v, off, 32);
        const int   oi = __shfl_xor(i, off, 32);
        if (ov > v || (ov == v && oi < i)) { v = ov; i = oi; }
      }
      if (lane == 0) { snext = i; idxOut[(size_t)b * M_ + s] = i; }
    }
    __syncthreads();
    const int nx = snext;
    if (t == (nx & 1023)) {
      const int kk = nx >> 10;
      float sx = 0.f, sy = 0.f, st = 0.f;
#pragma unroll
      for (int k = 0; k < 16; ++k)
        if (k == kk) { sx = px[k]; sy = py[k]; st = pt[k]; }   // selects, no spill
      slast[0] = sx; slast[1] = sy; slast[2] = st;
    }
    __syncthreads();
    lx = slast[0]; ly = slast[1]; lt = slast[2];
  }
}

// ------------------------------------------------------------------
// Kernel 2: fused gather + (W x + b) via V_WMMA_F32_16X16X4_F32 + LayerNorm.
// One 256-thread block per (batch, 16-row tile). 8 waves x 4 n-tiles = 512 cols.
// ------------------------------------------------------------------
__global__ __launch_bounds__(256) void proj_ln_kernel(
    const float* __restrict__ features, const int* __restrict__ idx,
    const float* __restrict__ W, const float* __restrict__ bias,
    const float* __restrict__ gamma, const float* __restrict__ beta,
    float* __restrict__ out)
{
  __shared__ float smem[16 * LDA];   // A tile (16 x 512, stride 516); reused as out tile (16 x 512)

  const int t    = threadIdx.x;
  const int lane = t & 31;
  const int wv   = t >> 5;
  const int tile = blockIdx.x;
  const int b    = tile >> 8;        // 256 tiles per batch
  const int m0   = (tile & 255) << 4;

  // ---- stage 16 gathered feature rows into LDS (float4, coalesced) ----
  {
    const int r  = t >> 4;
    const int cg = (t & 15) << 5;    // 32 floats per thread
    const int gi = idx[b * M_ + m0 + r];
    const float4* src = (const float4*)(features + ((size_t)b * N_ + gi) * D_ + cg);
    float4* dst = (float4*)(smem + r * LDA + cg);
#pragma unroll
    for (int i = 0; i < 8; ++i) dst[i] = src[i];
  }
  __syncthreads();

  // A operand layout (16x4 f32): lanes 0-15 -> row=lane, K k0..k0+1; lanes 16-31 -> row=lane-16, K k0+2..k0+3
  const int laneLo = lane & 15;
  const int khalf  = (lane >> 4) << 1;          // 0 or 2
  const float* arow = smem + laneLo * LDA + khalf;

  v8f acc[4];
#pragma unroll
  for (int j = 0; j < 4; ++j) {
    v8f c = {};
    const int n0 = (((j << 3) + wv) << 4);      // this wave's 16-col tile
    const float* wrow = W + (size_t)(n0 + laneLo) * D_ + khalf;  // B[k][n] = W[n][k]
#pragma unroll 8
    for (int k0 = 0; k0 < D_; k0 += 4) {
      v2f a  = *(const v2f*)(arow + k0);
      v2f bb = *(const v2f*)(wrow + k0);
      c = __builtin_amdgcn_wmma_f32_16x16x4_f32(false, a, false, bb, (short)0, c,
                                                false, false);
    }
    acc[j] = c;
  }
  __syncthreads();   // all waves done reading A -> reuse smem as output tile

  // D layout: lanes 0-15: VGPR v -> (m=v, n=lane); lanes 16-31: (m=8+v, n=lane-16)
  const int mhalf = (lane >> 4) << 3;
#pragma unroll
  for (int j = 0; j < 4; ++j) {
    const int n  = (((j << 3) + wv) << 4) + laneLo;
    const float bn = bias[n];
#pragma unroll
    for (int v = 0; v < 8; ++v)
      smem[(mhalf + v) * D_ + n] = acc[j][v] + bn;
  }
  __syncthreads();

  // ---- layernorm: wave wv handles rows 2*wv and 2*wv+1 (two-pass) ----
#pragma unroll
  for (int rr = 0; rr < 2; ++rr) {
    const int m = (wv << 1) + rr;
    const float* row = smem + m * D_;
    float s = 0.f;
#pragma unroll
    for (int i = 0; i < 16; ++i) s += row[lane + (i << 5)];
#pragma unroll
    for (int off = 16; off > 0; off >>= 1) s += __shfl_xor(s, off, 32);
    const float mu = s * (1.f / 512.f);
    float vs = 0.f;
#pragma unroll
    for (int i = 0; i < 16; ++i) { const float d = row[lane + (i << 5)] - mu; vs += d * d; }
#pragma unroll
    for (int off = 16; off > 0; off >>= 1) vs += __shfl_xor(vs, off, 32);
    const float rstd = rsqrtf(vs * (1.f / 512.f) + 1e-5f);
    float* orow = out + ((size_t)b * M_ + m0 + m) * D_;
#pragma unroll
    for (int i = 0; i < 16; ++i) {
      const int n = lane + (i << 5);
      orow[n] = (row[n] - mu) * rstd * gamma[n] + beta[n];
    }
  }
}

// ------------------------------------------------------------------
// Kernel 3: gather coords/times/polarities for the sampled indices.
// ------------------------------------------------------------------
__global__ __launch_bounds__(256) void gather_small_kernel(
    const float* __restrict__ coords, const float* __restrict__ times,
    const float* __restrict__ pol, const int* __restrict__ idx,
    float* __restrict__ coords_out, float* __restrict__ times_out,
    float* __restrict__ pol_out)
{
  const int tid = blockIdx.x * blockDim.x + threadIdx.x;
  if (tid >= B_ * M_) return;
  const int b  = tid >> 12;           // M_ = 4096
  const int gi = idx[tid];
  const size_t src = (size_t)b * N_ + gi;
  coords_out[2 * tid]     = coords[2 * src];
  coords_out[2 * tid + 1] = coords[2 * src + 1];
  times_out[tid] = times[src];
  pol_out[tid]   = pol[src];
}

extern "C" void kernel_launch(void* const* d_in, const int* in_sizes, int n_in,
                              void* d_out, int out_size, void* d_ws, size_t ws_size,
                              hipStream_t stream) {
  const float* features = (const float*)d_in[0];
  const float* coords   = (const float*)d_in[1];
  const float* times    = (const float*)d_in[2];
  const float* pol      = (const float*)d_in[3];
  const float* W        = (const float*)d_in[4];
  const float* bias     = (const float*)d_in[5];
  const float* gamma    = (const float*)d_in[6];
  const float* beta     = (const float*)d_in[7];

  float* out = (float*)d_out;
  int*   idx = (int*)d_ws;            // B_*M_ ints = 128 KB scratch

  fps_kernel<<<B_, 1024, 0, stream>>>(coords, times, idx);

  proj_ln_kernel<<<B_ * (M_ / 16), 256, 0, stream>>>(features, idx, W, bias,
                                                     gamma, beta, out);

  float* coords_out = out + (size_t)B_ * M_ * D_;
  float* times_out  = coords_out + (size_t)B_ * M_ * 2;
  float* pol_out    = times_out + (size_t)B_ * M_;
  gather_small_kernel<<<(B_ * M_ + 255) / 256, 256, 0, stream>>>(
      coords, times, pol, idx, coords_out, times_out, pol_out);
}